// FFF_Sparse_43499428774598
// MI455X (gfx1250) — compile-verified
//
#include <hip/hip_runtime.h>
#include <math.h>

typedef unsigned int u32;
typedef unsigned long long u64;
typedef u32 v4u __attribute__((ext_vector_type(4)));
typedef int  v8i __attribute__((ext_vector_type(8)));
typedef int  v4i __attribute__((ext_vector_type(4)));

#define FFF_D      768     // d_in == d_out
#define FFF_LVLS   12      // depth+1
#define FFF_VEC    6       // 6 float4 per lane: 6*4*32 = 768
#define WAVES_PB   4       // 4 waves (samples) per block -> 48 KB LDS

// Low 32 bits of a generic pointer into LDS == DS byte offset (flat LDS aperture
// maps addr[31:0] to LDS).
__device__ __forceinline__ u32 lds_byte_off(const void* p) {
    return (u32)(u64)p;
}

// Issue one TDM load of `nelem` f32 from global `gsrc` into LDS byte offset
// `lds_byte`. Descriptor per CDNA5 ISA section 8 (D# groups). Wave-uniform
// fields are forced through readfirstlane (TDM descriptors live in SGPRs).
__device__ __forceinline__ void tdm_load_row_f32(const float* gsrc, u32 lds_byte, int nelem) {
    u64 ga    = (u64)(const void*)gsrc;
    u32 ga_lo = (u32)__builtin_amdgcn_readfirstlane((int)(u32)ga);
    u32 ga_hi = (u32)__builtin_amdgcn_readfirstlane((int)((u32)(ga >> 32) & 0x01FFFFFFu));
    u32 lds_u = (u32)__builtin_amdgcn_readfirstlane((int)lds_byte);

    // Group 0 (128b): [1:0]=count=1 | [63:32]=lds_addr | [120:64]=global_addr | [127:126]=type=2
    v4u g0 = { 1u, lds_u, ga_lo, ga_hi | 0x80000000u };

    // Group 1 (256b):
    //  d0: workgroup_mask=0, data_size=2 (4B) at [17:16], no barrier/iterate/pad
    //  d1: atomic_barrier_addr=0 | tensor_dim0[15:0]  << 16
    //  d2: tensor_dim0[31:16]    | tensor_dim1[15:0]  << 16   (tensor_dim1 = 1)
    //  d3: tensor_dim1[31:16]    | tile_dim0          << 16
    //  d4: tile_dim1 (=1)        | tile_dim2 (=0)     << 16
    //  d5: tensor_dim0_stride[31:0] (data_size units)
    //  d6,d7: tensor_dim1_stride = 0 (unused, no Z tile)
    u32 n = (u32)nelem;
    v8i g1 = { (int)0x20000u,
               (int)((n & 0xFFFFu) << 16),
               (int)((n >> 16) | (1u << 16)),
               (int)(n << 16),
               (int)1,
               (int)n,
               0, 0 };
    v4i z4 = { 0, 0, 0, 0 };
    v8i z8 = { 0, 0, 0, 0, 0, 0, 0, 0 };
    // clang-23 / therock-10.0 lane: 6-arg form (extra int32x8 group, zero-filled
    // here since this descriptor is <= 2D).
    __builtin_amdgcn_tensor_load_to_lds(g0, g1, z4, z4, z8, 0);
}

__global__ __launch_bounds__(WAVES_PB * 32)
void FFF_Sparse_kernel(const float* __restrict__ x,
                       const float* __restrict__ w1,   // [n_nodes, 768] row-major
                       const float* __restrict__ w2,   // [n_nodes, 768] row-major
                       float* __restrict__ out,
                       int B) {
    // [wave][level-parity pair][child 0/1][row]: 4*2*2*768*4B = 48 KB
    __shared__ __align__(16) float lbuf[WAVES_PB][2][2][FFF_D];

    const int lane = threadIdx.x & 31;
    const int wave = threadIdx.x >> 5;
    const int b    = blockIdx.x * WAVES_PB + wave;
    if (b >= B) return;

    // ---- x[b] resident in VGPRs: 6 x float4 per lane (24 VGPRs) ----
    const float* xrow = x + (size_t)b * FFF_D;
    float4 xv[FFF_VEC];
#pragma unroll
    for (int k = 0; k < FFF_VEC; ++k)
        xv[k] = *(const float4*)(xrow + lane * 4 + k * 128);

    float4 acc[FFF_VEC];
#pragma unroll
    for (int k = 0; k < FFF_VEC; ++k)
        acc[k] = make_float4(0.f, 0.f, 0.f, 0.f);

    int node = 0;
    int sel  = 0;

#pragma unroll
    for (int lvl = 0; lvl < FFF_LVLS; ++lvl) {
        const int pair = lvl & 1;

        // Speculative TDM prefetch of BOTH children rows for the next level
        // into the other LDS pair; overlaps with this level's dot product.
        if (lvl < FFF_LVLS - 1) {
            const int c0 = 2 * node + 1;
            u32 base = lds_byte_off(&lbuf[wave][pair ^ 1][0][0]);
            tdm_load_row_f32(w1 + (size_t)c0 * FFF_D,       base,               FFF_D);
            tdm_load_row_f32(w1 + (size_t)(c0 + 1) * FFF_D, base + FFF_D * 4u,  FFF_D);
        }

        // ---- routing dot: <x[b], W1[node]> ----
        float partial = 0.f;
        if (lvl == 0) {
            // Root row straight from global (L2-resident).
            const float* r = w1;  // node 0
#pragma unroll
            for (int k = 0; k < FFF_VEC; ++k) {
                float4 wv = *(const float4*)(r + lane * 4 + k * 128);
                partial += xv[k].x * wv.x + xv[k].y * wv.y +
                           xv[k].z * wv.z + xv[k].w * wv.w;
            }
        } else {
            // Chosen child row already in LDS from previous level's TDM.
            const float* r = &lbuf[wave][pair][sel][0];
#pragma unroll
            for (int k = 0; k < FFF_VEC; ++k) {
                float4 wv = *(const float4*)(r + lane * 4 + k * 128);
                partial += xv[k].x * wv.x + xv[k].y * wv.y +
                           xv[k].z * wv.z + xv[k].w * wv.w;
            }
        }
        // wave32 butterfly reduction -> logit broadcast to all lanes
#pragma unroll
        for (int m = 16; m > 0; m >>= 1)
            partial += __shfl_xor(partial, m, 32);
        const float logit = partial;

        // ---- fused epilogue: acc += gelu(logit) * W2[node] ----
        const float a = 0.5f * logit * (1.f + erff(logit * 0.70710678118654752f));
        const float* w2row = w2 + (size_t)node * FFF_D;
#pragma unroll
        for (int k = 0; k < FFF_VEC; ++k) {
            float4 wv = *(const float4*)(w2row + lane * 4 + k * 128);
            acc[k].x += a * wv.x; acc[k].y += a * wv.y;
            acc[k].z += a * wv.z; acc[k].w += a * wv.w;
        }

        // ---- descend ----
        const int bit = (logit >= 0.f) ? 1 : 0;
        sel  = bit;
        node = 2 * node + 1 + bit;

        // TDM completion fence before next level reads the prefetched LDS pair.
        if (lvl < FFF_LVLS - 1)
            __builtin_amdgcn_s_wait_tensorcnt(0);
    }

    float* orow = out + (size_t)b * FFF_D;
#pragma unroll
    for (int k = 0; k < FFF_VEC; ++k)
        *(float4*)(orow + lane * 4 + k * 128) = acc[k];
}

extern "C" void kernel_launch(void* const* d_in, const int* in_sizes, int n_in,
                              void* d_out, int out_size, void* d_ws, size_t ws_size,
                              hipStream_t stream) {
    const float* x   = (const float*)d_in[0];
    const float* w1s = (const float*)d_in[1];
    const float* w2s = (const float*)d_in[2];
    float* out = (float*)d_out;

    const int B = in_sizes[0] / FFF_D;  // 16384
    const int blocks = (B + WAVES_PB - 1) / WAVES_PB;
    FFF_Sparse_kernel<<<blocks, WAVES_PB * 32, 0, stream>>>(x, w1s, w2s, out, B);
}